// MultiTaskLoss_84035330114062
// MI455X (gfx1250) — compile-verified
//
#include <hip/hip_runtime.h>
#include <math.h>

#define BNP 128   // num pred queries
#define BNG 64    // num gt

typedef __attribute__((ext_vector_type(2))) float v2f;
typedef __attribute__((ext_vector_type(8))) float v8f;

__device__ __forceinline__ unsigned int f32_ordkey(float f) {
    unsigned int u = __float_as_uint(f);
    return (u & 0x80000000u) ? ~u : (u | 0x80000000u);
}
__device__ __forceinline__ float f32_ordkey_inv(unsigned int k) {
    unsigned int u = (k & 0x80000000u) ? (k ^ 0x80000000u) : ~k;
    return __uint_as_float(u);
}

// ---------------------------------------------------------------------------
// Kernel 1: cost[b][m][n] = || pred[b][m] - gt[b][n] ||_2  via WMMA f32 16x16x4
// One wave (32 lanes) per 16x16 tile.  grid.x = B * 8 * 4, block = 32.
// ---------------------------------------------------------------------------
__global__ void cost_kernel(const float* __restrict__ pred,
                            const float* __restrict__ gt,
                            float* __restrict__ cost) {
    int t    = blockIdx.x;
    int tn   = t & 3;          // gt-tile   0..3
    int tm   = (t >> 2) & 7;   // pred-tile 0..7
    int b    = t >> 5;         // sample
    int lane = threadIdx.x;    // 0..31 (wave32)
    int half = lane >> 4;      // 0: K=0,1   1: K=2,3
    int l16  = lane & 15;
    int k0   = 2 * half;

    // A matrix 16x4 (M=pred rows, K=coord dim padded to 4)
    int m = tm * 16 + l16;
    const float* prow = pred + ((size_t)b * BNP + m) * 3;
    float p0 = prow[0], p1 = prow[1], p2 = prow[2];
    v2f a;
    a.x = half ? p2 : p0;
    a.y = half ? 0.0f : p1;

    // B matrix 4x16 (K=coord dim, N=gt cols):  B[k][n] = gt[n][k]
    int n = tn * 16 + l16;
    const float* grow = gt + ((size_t)b * BNG + n) * 3;
    float g0 = grow[0], g1 = grow[1], g2 = grow[2];
    v2f bm;
    bm.x = half ? g2 : g0;
    bm.y = half ? 0.0f : g1;

    v8f c = {};
    // D = A x B : dot products pred_m . gt_n
    c = __builtin_amdgcn_wmma_f32_16x16x4_f32(false, a, false, bm,
                                              (short)0, c, false, false);

    float gn2     = g0 * g0 + g1 * g1 + g2 * g2;  // this lane's column norm
    float pn2self = p0 * p0 + p1 * p1 + p2 * p2;  // norm of row tm*16+l16

    // D layout: VGPR r on lane L -> row (r + 8*half), col l16
    float* crow = cost + ((size_t)b * BNP + tm * 16 + 8 * half) * BNG + n;
#pragma unroll
    for (int r = 0; r < 8; ++r) {
        float pn2 = __shfl(pn2self, r + 8 * half, 32);  // row norm via lane bcast
        float d2  = pn2 + gn2 - 2.0f * c[r];
        d2 = d2 > 0.0f ? d2 : 0.0f;
        crow[(size_t)r * BNG] = sqrtf(d2);
    }
}

// ---------------------------------------------------------------------------
// Kernel 2: Jonker-Volgenant assignment, one block (128 threads = 4 waves)
// per sample.  Transposed problem: n = BNG rows (gt), m = BNP cols (pred).
// Ct[i][j] = cost[pred j-1][gt i-1]  (1-based i,j).
// Output: match[b][pred] = matched gt index, or -1.
// ---------------------------------------------------------------------------
__global__ void hungarian_kernel(const float* __restrict__ cost,
                                 int* __restrict__ match) {
    const int b   = blockIdx.x;
    const int tid = threadIdx.x;          // 0..127; owns column j = tid+1
    const int j   = tid + 1;
    const int wid = tid >> 5;             // wave id 0..3
    const int lid = tid & 31;
    const float* C = cost + (size_t)b * BNP * BNG;

    __shared__ float u[BNG + 1];
    __shared__ float v[BNP + 1];
    __shared__ int   p[BNP + 1];
    __shared__ int   way[BNP + 1];
    __shared__ float minv[BNP + 1];
    __shared__ int   used[BNP + 1];
    __shared__ unsigned long long wred[4];
    __shared__ float s_delta;
    __shared__ int   s_j0;

    if (tid <= BNG) u[tid] = 0.0f;
    v[j] = 0.0f; p[j] = 0; way[j] = 0;
    if (tid == 0) { v[0] = 0.0f; p[0] = 0; way[0] = 0; }
    __syncthreads();

    for (int i = 1; i <= BNG; ++i) {
        minv[j] = __builtin_inff();
        used[j] = 0;
        if (tid == 0) { p[0] = i; s_j0 = 0; used[0] = 0; minv[0] = __builtin_inff(); }
        __syncthreads();

        while (true) {
            int j0 = s_j0;
            if (tid == 0) used[j0] = 1;
            __syncthreads();

            int   i0 = p[j0];
            float mv = __builtin_inff();
            if (!used[j]) {
                float cur = C[(size_t)(j - 1) * BNG + (i0 - 1)] - u[i0] - v[j];
                if (cur < minv[j]) { minv[j] = cur; way[j] = j0; }
                mv = minv[j];
            }
            // (value, col) argmin: monotone u64 key, wave32 shuffle reduce,
            // then a single LDS combine across the 4 waves.
            unsigned long long key =
                ((unsigned long long)f32_ordkey(mv) << 32) | (unsigned int)j;
#pragma unroll
            for (int off = 16; off > 0; off >>= 1) {
                unsigned long long o = __shfl_xor(key, off, 32);
                key = (o < key) ? o : key;
            }
            if (lid == 0) wred[wid] = key;
            __syncthreads();
            if (tid == 0) {
                unsigned long long k = wred[0];
                if (wred[1] < k) k = wred[1];
                if (wred[2] < k) k = wred[2];
                if (wred[3] < k) k = wred[3];
                s_delta = f32_ordkey_inv((unsigned int)(k >> 32));
                s_j0    = (int)(k & 0xFFFFFFFFu);
            }
            __syncthreads();

            float delta = s_delta;
            // dual updates: matched rows of used columns are distinct -> race-free
            if (used[j]) { u[p[j]] += delta; v[j] -= delta; }
            else         { minv[j] -= delta; }
            if (tid == 0) { u[p[0]] += delta; v[0] -= delta; }  // j=0 always used
            __syncthreads();

            if (p[s_j0] == 0) break;   // uniform across block
        }

        if (tid == 0) {               // augment along alternating path
            int j0 = s_j0;
            while (j0) {
                int j1 = way[j0];
                p[j0]  = p[j1];
                j0     = j1;
            }
        }
        __syncthreads();
    }

    match[(size_t)b * BNP + tid] = p[j] ? (p[j] - 1) : -1;
}

// ---------------------------------------------------------------------------
// Kernel 3: per-sample coord smooth-L1 + weighted existence BCE.
// One block (128 threads = 4 waves) per sample; shuffle reductions.
// ---------------------------------------------------------------------------
__global__ void loss_kernel(const float* __restrict__ pred,
                            const float* __restrict__ gt,
                            const float* __restrict__ exist,
                            const float* __restrict__ weight,
                            const int* __restrict__ match,
                            float* __restrict__ per_sample) {
    const int b   = blockIdx.x;
    const int tid = threadIdx.x;
    const int wid = tid >> 5;
    const int lid = tid & 31;
    const int g   = match[(size_t)b * BNP + tid];

    float coord = 0.0f;
    if (g >= 0) {
        const float* p = pred + ((size_t)b * BNP + tid) * 3;
        const float* q = gt   + ((size_t)b * BNG + g)   * 3;
#pragma unroll
        for (int d = 0; d < 3; ++d) {
            float ad = fabsf(p[d] - q[d]);
            coord += (ad < 1.0f) ? 0.5f * ad * ad : (ad - 0.5f);
        }
    }
    float pe   = exist[(size_t)b * BNP + tid];
    float lp   = fmaxf(logf(pe), -100.0f);
    float l1p  = fmaxf(log1pf(-pe), -100.0f);
    float mask = (g >= 0) ? 1.0f : 0.0f;
    float bce  = -(mask * lp + (1.0f - mask) * l1p) * weight[tid];

#pragma unroll
    for (int off = 16; off > 0; off >>= 1) {
        coord += __shfl_xor(coord, off, 32);
        bce   += __shfl_xor(bce,   off, 32);
    }
    __shared__ float sc[4];
    __shared__ float sb[4];
    if (lid == 0) { sc[wid] = coord; sb[wid] = bce; }
    __syncthreads();
    if (tid == 0) {
        float c4 = sc[0] + sc[1] + sc[2] + sc[3];
        float b4 = sb[0] + sb[1] + sb[2] + sb[3];
        per_sample[b] = c4 / (float)(BNG * 3) + b4 / (float)BNP;
    }
}

// ---------------------------------------------------------------------------
// Kernel 4: reduce over batch + count MSE.  One wave32 block, lane per sample.
// ---------------------------------------------------------------------------
__global__ void final_kernel(const float* __restrict__ per_sample,
                             const float* __restrict__ count_pred,
                             const int* __restrict__ gt_counts,
                             float* __restrict__ out, int B) {
    const int tid = threadIdx.x;  // B == 32
    float ce   = (tid < B) ? per_sample[tid] : 0.0f;
    float diff = (tid < B) ? (count_pred[tid] - (float)gt_counts[tid]) : 0.0f;
    float cnt  = diff * diff;
#pragma unroll
    for (int off = 16; off > 0; off >>= 1) {
        ce  += __shfl_xor(ce,  off, 32);
        cnt += __shfl_xor(cnt, off, 32);
    }
    if (tid == 0) out[0] = ce / (float)B + cnt / (float)B;
}

// ---------------------------------------------------------------------------
extern "C" void kernel_launch(void* const* d_in, const int* in_sizes, int n_in,
                              void* d_out, int out_size, void* d_ws, size_t ws_size,
                              hipStream_t stream) {
    const float* pred       = (const float*)d_in[0];  // [B,128,3]
    const float* exist      = (const float*)d_in[1];  // [B,128]
    const float* count_pred = (const float*)d_in[2];  // [B,1]
    const float* gt         = (const float*)d_in[3];  // [B,64,3]
    const int*   gt_counts  = (const int*)d_in[4];    // [B]
    const float* weight     = (const float*)d_in[5];  // [1,128]
    float*       out        = (float*)d_out;

    const int B = in_sizes[4];

    float* cost       = (float*)d_ws;
    int*   match      = (int*)((char*)d_ws + (size_t)B * BNP * BNG * sizeof(float));
    float* per_sample = (float*)((char*)match + (size_t)B * BNP * sizeof(int));

    cost_kernel<<<B * 8 * 4, 32, 0, stream>>>(pred, gt, cost);
    hungarian_kernel<<<B, BNP, 0, stream>>>(cost, match);
    loss_kernel<<<B, BNP, 0, stream>>>(pred, gt, exist, weight, match, per_sample);
    final_kernel<<<1, 32, 0, stream>>>(per_sample, count_pred, gt_counts, out, B);
}